// CIFModule_70188355551472
// MI455X (gfx1250) — compile-verified
//
#include <hip/hip_runtime.h>
#include <hip/hip_bf16.h>
#include <math.h>

// ---------------- problem constants (from reference setup_inputs) ----------
#define B_DIM   16
#define T_LEN   3000
#define C_DIM   32
#define H_DIM   32
#define W_DIM   375
#define DSW     192
#define N_FIRES 128
#define P_DIM   32
#define DP      32
#define DM      512
#define HID     128

#define ACOUSTIC_ELEMS ((long)B_DIM * N_FIRES * (P_DIM + H_DIM) * DM) // 67,108,864
#define ALPHA_OFF      ACOUSTIC_ELEMS
#define QTY_OFF        (ACOUSTIC_ELEMS + (long)B_DIM * T_LEN)

typedef __attribute__((ext_vector_type(2))) float v2f;
typedef __attribute__((ext_vector_type(8))) float v8f;

// ---------------------------------------------------------------------------
// Kernel 1: alpha = softplus(proj(relu(dense(LN(dwconv(x)+x))))); fire bits;
// wave-reduced per-batch onset_prob sums (atomic). One thread per (b,t).
// ---------------------------------------------------------------------------
__global__ __launch_bounds__(256) void k_alpha(
    const float* __restrict__ fs, const float* __restrict__ conv_w,
    const float* __restrict__ ln_w, const float* __restrict__ ln_b,
    const float* __restrict__ dense_w, const float* __restrict__ dense_b,
    const float* __restrict__ proj_w, const float* __restrict__ proj_b,
    float* __restrict__ alpha_out, int* __restrict__ fires,
    float* __restrict__ sums)
{
    __shared__ float s_cw[C_DIM * 3];
    __shared__ float s_lw[C_DIM], s_lb[C_DIM];
    __shared__ float s_dw[C_DIM * HID];
    __shared__ float s_db[HID], s_pw[HID];

    const int tid = threadIdx.x;
    for (int i = tid; i < C_DIM * 3; i += 256) s_cw[i] = conv_w[i];
    if (tid < C_DIM) { s_lw[tid] = ln_w[tid]; s_lb[tid] = ln_b[tid]; }
    for (int i = tid; i < C_DIM * HID; i += 256) s_dw[i] = dense_w[i];
    if (tid < HID) { s_db[tid] = dense_b[tid]; s_pw[tid] = proj_w[tid]; }
    __syncthreads();

    const int idx = blockIdx.x * 256 + tid;
    const bool act = idx < B_DIM * T_LEN;
    const int b = act ? idx / T_LEN : 0;
    const int t = act ? idx % T_LEN : 0;

    float onset = 0.f;
    if (act) {
        const float* base = fs + (size_t)idx * C_DIM;
        float x[C_DIM];
        float mu = 0.f;
#pragma unroll
        for (int c = 0; c < C_DIM; ++c) {
            float ctr = base[c];
            float lft = (t > 0)         ? base[c - C_DIM] : 0.f;
            float rgt = (t < T_LEN - 1) ? base[c + C_DIM] : 0.f;
            // cross-correlation (XLA conv), kernel 3, SAME zero pad; +residual
            float v = ctr + s_cw[c*3+0]*lft + s_cw[c*3+1]*ctr + s_cw[c*3+2]*rgt;
            x[c] = v; mu += v;
        }
        mu *= (1.f / C_DIM);
        float var = 0.f;
#pragma unroll
        for (int c = 0; c < C_DIM; ++c) { float d = x[c] - mu; var += d * d; }
        var *= (1.f / C_DIM);
        const float rs = rsqrtf(var + 1e-5f);
#pragma unroll
        for (int c = 0; c < C_DIM; ++c)
            x[c] = (x[c] - mu) * rs * s_lw[c] + s_lb[c];

        float pa = proj_b[0];
        for (int h = 0; h < HID; ++h) {
            float acc = s_db[h];
#pragma unroll
            for (int c = 0; c < C_DIM; ++c) acc += x[c] * s_dw[c * HID + h];
            acc = fmaxf(acc, 0.f);
            pa += acc * s_pw[h];
        }
        // softplus (stable)
        float alpha = (pa > 20.f) ? pa : log1pf(__expf(pa));
        alpha_out[idx] = alpha;
        fires[idx] = (alpha > 1.0f) ? 1 : 0;
        onset = 1.f / (1.f + __expf(-(alpha - 1.0f) * 10.f)); // sigmoid((a-1)/0.1)
    }

    // wave32 reduction of onset by batch (block spans at most 2 batches)
    const int lane = tid & 31;
    const int bfirst = __shfl(b, 0, 32);
    float v = (b == bfirst) ? onset : 0.f;
#pragma unroll
    for (int o = 16; o > 0; o >>= 1) v += __shfl_down(v, o, 32);
    if (lane == 0) atomicAdd(&sums[bfirst], v);
    if (act && b != bfirst) atomicAdd(&sums[b], onset);
}

// ---------------------------------------------------------------------------
// Kernel 2: per-batch sequential scan == cumsum + searchsorted; emit fire_w
// (time bin mapped to swin column). Thread 0 also finishes qty_loss.
// ---------------------------------------------------------------------------
__global__ void k_fires(const int* __restrict__ fires,
                        const float* __restrict__ sums,
                        const int* __restrict__ tgt,
                        int* __restrict__ fire_w,
                        float* __restrict__ qty_out)
{
    const int b = threadIdx.x;
    if (b < B_DIM) {
        const int* f = fires + (size_t)b * T_LEN;
        const int base = b * N_FIRES;
        int cnt = 0;
        for (int t = 0; t < T_LEN; ++t) {
            if (f[t]) {
                if (cnt < N_FIRES) {
                    int w = (int)((float)t * (float)W_DIM / (float)T_LEN);
                    fire_w[base + cnt] = (w > W_DIM - 1) ? (W_DIM - 1) : w;
                }
                ++cnt;
            }
        }
        for (int n = (cnt < N_FIRES ? cnt : N_FIRES); n < N_FIRES; ++n)
            fire_w[base + n] = 374; // searchsorted miss -> min(T, T-1)=2999 -> col 374
    }
    if (threadIdx.x == 0) {
        float q = 0.f;
        for (int i = 0; i < B_DIM; ++i) q += fabsf(sums[i] - (float)tgt[i]);
        *qty_out = q * (1.f / B_DIM);
    }
}

// ---------------------------------------------------------------------------
// f32 WMMA helper fragments:
//   A lane L: a[j] = A[L%16][2*(L/16)+j]
//   B lane L: b[j] = B[2*(L/16)+j][L%16]
//   C/D VGPR v, lane L: (m = v + 8*(L/16), n = L%16)
// ---------------------------------------------------------------------------

// Kernel 3: pitch_tok = pitch_tokens @ pitch_w + pitch_b
// rows = B*N*P = 65536 (contiguous, k=32), cols = 512.
// grid (4096 M-tiles, 4) x 256 threads; wave w -> N-tile (gy*8+w).
__global__ __launch_bounds__(256) void k_pitch(
    const float* __restrict__ A, const float* __restrict__ Wt,
    const float* __restrict__ bias, float* __restrict__ out)
{
    __shared__ float sA[16 * 34];           // pad 32 -> 34 (bank-conflict free)
    const int tid = threadIdx.x;
    const long m0 = (long)blockIdx.x * 16;

    { // stage 16x32 A tile (contiguous 512 floats) as float2
        const int row = tid >> 4, k2 = tid & 15;
        const float2* g = (const float2*)(A + m0 * DP);
        float2 v = g[row * 16 + k2];
        sA[row * 34 + k2 * 2]     = v.x;
        sA[row * 34 + k2 * 2 + 1] = v.y;
    }
    __syncthreads();

    const int wave = tid >> 5, lane = tid & 31;
    const int nIdx = lane & 15, khalf = lane >> 4;
    const int n0 = (blockIdx.y * 8 + wave) * 16;

    const float bv = bias[n0 + nIdx];
    v8f acc = {bv, bv, bv, bv, bv, bv, bv, bv};

    const float* arow = sA + nIdx * 34;
#pragma unroll
    for (int k0 = 0; k0 < DP / 4; ++k0) {
        const int kk = k0 * 4 + khalf * 2;
        v2f a; a[0] = arow[kk]; a[1] = arow[kk + 1];
        v2f bb; bb[0] = Wt[(size_t)kk * DM + n0 + nIdx];
                bb[1] = Wt[(size_t)(kk + 1) * DM + n0 + nIdx];
        acc = __builtin_amdgcn_wmma_f32_16x16x4_f32(
            false, a, false, bb, (short)0, acc, false, false);
    }

    // all 16 rows of this tile share (b,n); p0 in {0,16}
    const int b  = (int)(m0 >> 12);            // / (N*P)
    const int n  = (int)((m0 >> 5) & 127);
    const int p0 = (int)(m0 & 31);
    float* obase = out + (((long)(b * N_FIRES + n) * 64 + p0) * DM) + n0 + nIdx;
#pragma unroll
    for (int v = 0; v < 8; ++v)
        obase[(long)(v + 8 * khalf) * DM] = acc[v];
}

// Kernel 4: swin_tok = gather(swin_2d at fire_w) @ swin_w + swin_b
// rows = (b,n,h) = 65536, k = 192, cols = 512. A-rows gathered via fire_w.
__global__ __launch_bounds__(256) void k_swin(
    const float* __restrict__ swin, const float* __restrict__ Wt,
    const float* __restrict__ bias, const int* __restrict__ fire_w,
    float* __restrict__ out)
{
    __shared__ float sA[16 * 194];          // pad 192 -> 194 (bank-conflict free)
    const int tid = threadIdx.x;
    const long g0 = (long)blockIdx.x * 16;
    const int b     = (int)(g0 >> 12);      // / (N*H)
    const int n     = (int)((g0 >> 5) & 127);
    const int hbase = (int)(g0 & 31);       // 0 or 16
    const int fw    = fire_w[b * N_FIRES + n];

    // stage gathered 16x192 A tile; 1536 float2 / 256 threads = 6 each
#pragma unroll
    for (int j = 0; j < 6; ++j) {
        const int i2  = tid + 256 * j;
        const int row = i2 / 96, k2 = i2 % 96;
        const float* rp =
            swin + ((long)(b * H_DIM + hbase + row) * W_DIM + fw) * DSW;
        float2 v = *(const float2*)(rp + k2 * 2);
        sA[row * 194 + k2 * 2]     = v.x;
        sA[row * 194 + k2 * 2 + 1] = v.y;
    }
    __syncthreads();

    const int wave = tid >> 5, lane = tid & 31;
    const int nIdx = lane & 15, khalf = lane >> 4;
    const int n0 = (blockIdx.y * 8 + wave) * 16;

    const float bv = bias[n0 + nIdx];
    v8f acc = {bv, bv, bv, bv, bv, bv, bv, bv};

    const float* arow = sA + nIdx * 194;
#pragma unroll 4
    for (int k0 = 0; k0 < DSW / 4; ++k0) {
        const int kk = k0 * 4 + khalf * 2;
        v2f a; a[0] = arow[kk]; a[1] = arow[kk + 1];
        v2f bb; bb[0] = Wt[(size_t)kk * DM + n0 + nIdx];
                bb[1] = Wt[(size_t)(kk + 1) * DM + n0 + nIdx];
        acc = __builtin_amdgcn_wmma_f32_16x16x4_f32(
            false, a, false, bb, (short)0, acc, false, false);
    }

    float* obase =
        out + (((long)(b * N_FIRES + n) * 64 + P_DIM + hbase) * DM) + n0 + nIdx;
#pragma unroll
    for (int v = 0; v < 8; ++v)
        obase[(long)(v + 8 * khalf) * DM] = acc[v];
}

// ---------------------------------------------------------------------------
extern "C" void kernel_launch(void* const* d_in, const int* in_sizes, int n_in,
                              void* d_out, int out_size, void* d_ws, size_t ws_size,
                              hipStream_t stream) {
    const float* fire_signal  = (const float*)d_in[0];
    const float* swin_2d      = (const float*)d_in[1];
    const float* pitch_tokens = (const float*)d_in[2];
    const int*   target_len   = (const int*)  d_in[3];
    const float* conv_w       = (const float*)d_in[4];
    const float* ln_w         = (const float*)d_in[5];
    const float* ln_b         = (const float*)d_in[6];
    const float* dense_w      = (const float*)d_in[7];
    const float* dense_b      = (const float*)d_in[8];
    const float* proj_w       = (const float*)d_in[9];
    const float* proj_b       = (const float*)d_in[10];
    const float* pitch_w      = (const float*)d_in[11];
    const float* pitch_b      = (const float*)d_in[12];
    const float* swin_w       = (const float*)d_in[13];
    const float* swin_b       = (const float*)d_in[14];

    float* out = (float*)d_out;
    float* alpha_out = out + ALPHA_OFF;
    float* qty_out   = out + QTY_OFF;

    // workspace: [0,64) onset sums (16 f32), [64, 64+192000) fire bits,
    // [192064, +8192) fire_w ints
    char* ws = (char*)d_ws;
    float* sums   = (float*)ws;
    int*   fires  = (int*)(ws + 64);
    int*   fire_w = (int*)(ws + 64 + (size_t)B_DIM * T_LEN * 4);

    hipMemsetAsync(sums, 0, 64, stream);

    const int nPos = B_DIM * T_LEN;
    k_alpha<<<(nPos + 255) / 256, 256, 0, stream>>>(
        fire_signal, conv_w, ln_w, ln_b, dense_w, dense_b, proj_w, proj_b,
        alpha_out, fires, sums);

    k_fires<<<1, 32, 0, stream>>>(fires, sums, target_len, fire_w, qty_out);

    k_pitch<<<dim3(4096, 4), 256, 0, stream>>>(pitch_tokens, pitch_w, pitch_b, out);

    k_swin<<<dim3(4096, 4), 256, 0, stream>>>(swin_2d, swin_w, swin_b, fire_w, out);
}